// Scatter__36266703848187
// MI455X (gfx1250) — compile-verified
//
#include <hip/hip_runtime.h>
#include <hip/hip_bf16.h>

// Batched scatter-add:  out[b, vs[j], :] += x[b, j, :],  out zero elsewhere.
// B=256, NSEL=60000, NVERTS=100000, F=3, fp32.
//
// Strategy: invert the scatter into a gather via CSR inversion of `vs`
// (histogram -> exclusive scan -> permutation fill), then one thread per
// (batch, vertex) row sums its segment and writes the output row exactly
// once. Output rows are staged per-wave in LDS and emitted with CDNA5
// GLOBAL_STORE_ASYNC_FROM_LDS_B128 (async LDS->memory engine, ASYNCcnt),
// giving fully-coalesced 16B stores with NT hint. This achieves the
// read-once/write-once HBM minimum (~0.49 GB -> ~21 us at 23.3 TB/s) and
// avoids float atomics entirely in the hot path.

#define THREADS 256
#define SCAN_THREADS 1024

__global__ void zero_i32_kernel(int* __restrict__ p, int n) {
    int i = blockIdx.x * blockDim.x + threadIdx.x;
    if (i < n) p[i] = 0;
}

__global__ void count_kernel(const int* __restrict__ vs,
                             int* __restrict__ counts, int nsel) {
    int j = blockIdx.x * blockDim.x + threadIdx.x;
    if (j < nsel) atomicAdd(&counts[vs[j]], 1);
}

// Single-workgroup exclusive scan over `nverts` counts.
// Writes offsets[0..nverts] and turns `counts` into a cursor copy of offsets.
// Index data is ~0.8 MB total -> L2-resident, cost is negligible.
__global__ __launch_bounds__(SCAN_THREADS)
void scan_kernel(int* __restrict__ counts, int* __restrict__ offsets, int nverts) {
    __shared__ int ssum[SCAN_THREADS];
    const int tid   = threadIdx.x;
    const int chunk = (nverts + SCAN_THREADS - 1) / SCAN_THREADS;
    const int base  = tid * chunk;
    const int lim   = (base + chunk < nverts) ? (base + chunk) : nverts;

    int s = 0;
    for (int i = base; i < lim; ++i) s += counts[i];
    ssum[tid] = s;
    __syncthreads();

    // Hillis-Steele inclusive scan across thread partial sums.
    for (int off = 1; off < SCAN_THREADS; off <<= 1) {
        int v = (tid >= off) ? ssum[tid - off] : 0;
        __syncthreads();
        ssum[tid] += v;
        __syncthreads();
    }

    int run = ssum[tid] - s;  // exclusive prefix for this thread's chunk
    for (int i = base; i < lim; ++i) {
        int c = counts[i];
        offsets[i] = run;
        counts[i]  = run;   // cursor for the fill pass
        run += c;
    }
    if (tid == SCAN_THREADS - 1) offsets[nverts] = ssum[SCAN_THREADS - 1];
}

__global__ void fill_kernel(const int* __restrict__ vs,
                            int* __restrict__ cursor,
                            int* __restrict__ perm, int nsel) {
    int j = blockIdx.x * blockDim.x + threadIdx.x;
    if (j < nsel) {
        int p = atomicAdd(&cursor[vs[j]], 1);
        perm[p] = j;
    }
}

struct F3 { float a, b, c; };  // 12-byte row -> global_load_b96

// One thread per (b, v) output row. Reads are gathered 12B rows from the
// L2-resident x[b] slice (720 KB); each x element is read exactly once
// overall. Each wave stages its 32 rows (384 contiguous bytes) in LDS and
// issues 24 GLOBAL_STORE_ASYNC_FROM_LDS_B128 ops (wave-local: needs only
// s_wait_dscnt, no barrier). Async completion is covered by the implicit
// S_WAIT_IDLE at S_ENDPGM.
__global__ __launch_bounds__(THREADS)
void gather_kernel(const float* __restrict__ x,
                   const int* __restrict__ offsets,
                   const int* __restrict__ perm,
                   float* __restrict__ out,
                   int nverts, int nsel) {
    __shared__ float tile[THREADS * 3];  // 3072 B, LDS offset 0 (only shared obj)

    const int tid = threadIdx.x;
    const int v0  = blockIdx.x * THREADS;
    const int v   = v0 + tid;
    const int b   = blockIdx.y;

    float a0 = 0.f, a1 = 0.f, a2 = 0.f;
    if (v < nverts) {
        int o0 = offsets[v];
        int o1 = offsets[v + 1];
        const float* xb = x + (size_t)b * (size_t)nsel * 3u;
        for (int k = o0; k < o1; ++k) {
            int j = perm[k];
            F3 r = *(const F3*)(xb + (size_t)j * 3u);
            a0 += r.a; a1 += r.b; a2 += r.c;
        }
    }

    // Stage this block's rows in LDS, row-major: tile[tid*3 + i].
    tile[tid * 3 + 0] = a0;
    tile[tid * 3 + 1] = a1;
    tile[tid * 3 + 2] = a2;

    const int valid  = (nverts - v0 < THREADS) ? (nverts - v0) : THREADS;  // rows
    const int vbytes = valid * 12;
    const int full   = vbytes & ~15;          // 16B-aligned async-covered prefix

    const int wave  = tid >> 5;
    const int lane  = tid & 31;
    const int wbyte = wave * 384 + lane * 16; // byte offset within block tile

    // Wave's LDS writes must be visible to the async store engine.
    asm volatile("s_wait_dscnt 0x0" ::: "memory");

    const float* blockbase = out + ((size_t)b * (size_t)nverts + (size_t)v0) * 3u;

    if (lane < 24 && wbyte < full) {
        unsigned goff    = (unsigned)wbyte;   // byte offset from blockbase (GVS)
        unsigned lds_off = (unsigned)wbyte;   // tile lives at LDS offset 0
        asm volatile("global_store_async_from_lds_b128 %0, %1, %2 th:TH_STORE_NT"
                     :: "v"(goff), "v"(lds_off), "s"(blockbase)
                     : "memory");
    }

    // Tail rows whose 12B range is not fully covered by the async region:
    // store from registers (overlap with async region rewrites same values).
    if (v < nverts && (tid * 12 + 12) > full) {
        float* op = out + ((size_t)b * (size_t)nverts + (size_t)v) * 3u;
        __builtin_nontemporal_store(a0, op + 0);
        __builtin_nontemporal_store(a1, op + 1);
        __builtin_nontemporal_store(a2, op + 2);
    }
}

extern "C" void kernel_launch(void* const* d_in, const int* in_sizes, int n_in,
                              void* d_out, int out_size, void* d_ws, size_t ws_size,
                              hipStream_t stream) {
    const float* x  = (const float*)d_in[0];
    const int*   vs = (const int*)d_in[1];
    float*       out = (float*)d_out;

    const int B      = 256;
    const int F      = 3;
    const int NSEL   = in_sizes[1];
    const int NVERTS = out_size / (B * F);

    // Workspace layout (ints): counts/cursor[NVERTS] | offsets[NVERTS+1] | perm[NSEL]
    int* counts  = (int*)d_ws;
    int* offsets = counts + NVERTS;
    int* perm    = offsets + NVERTS + 1;

    zero_i32_kernel<<<(NVERTS + THREADS - 1) / THREADS, THREADS, 0, stream>>>(counts, NVERTS);
    count_kernel  <<<(NSEL   + THREADS - 1) / THREADS, THREADS, 0, stream>>>(vs, counts, NSEL);
    scan_kernel   <<<1, SCAN_THREADS, 0, stream>>>(counts, offsets, NVERTS);
    fill_kernel   <<<(NSEL   + THREADS - 1) / THREADS, THREADS, 0, stream>>>(vs, counts, perm, NSEL);

    dim3 grid((NVERTS + THREADS - 1) / THREADS, B);
    gather_kernel<<<grid, THREADS, 0, stream>>>(x, offsets, perm, out, NVERTS, NSEL);
}